// UnitaryPMM_61924838474104
// MI455X (gfx1250) — compile-verified
//
#include <hip/hip_runtime.h>
#include <hip/hip_bf16.h>

typedef float v2f __attribute__((ext_vector_type(2)));
typedef float v8f __attribute__((ext_vector_type(8)));

// ---------------------------------------------------------------------------
// Prep: hermitize H0 / Hl / S once into workspace, in WMMA-B-operand layout.
//   Bg  [64][80] : row l = (herm(Hl[l]).re flat36 | herm(Hl[l]).im flat36 | 0 pad)
//   H0f [80]     : herm(H0) features (re36|im36|pad)
//   W2  [72][32] : W2[p][k] = Sfeat so out[b,k] = f[b]·W2[:,k]
// ---------------------------------------------------------------------------
__global__ void UnitaryPMM_prep(const float* __restrict__ H0re, const float* __restrict__ H0im,
                                const float* __restrict__ Hlre, const float* __restrict__ Hlim,
                                const float* __restrict__ Sre,  const float* __restrict__ Sim,
                                float* __restrict__ Bg, float* __restrict__ H0f,
                                float* __restrict__ W2) {
  const int t = threadIdx.x;
  for (int idx = t; idx < 64 * 80; idx += 256) {
    int l = idx / 80, p = idx % 80;
    float v = 0.f;
    if (p < 36) { int i = p / 6, j = p % 6;
      v = 0.5f * (Hlre[l * 36 + i * 6 + j] + Hlre[l * 36 + j * 6 + i]); }
    else if (p < 72) { int q = p - 36; int i = q / 6, j = q % 6;
      v = 0.5f * (Hlim[l * 36 + i * 6 + j] - Hlim[l * 36 + j * 6 + i]); }
    Bg[idx] = v;
  }
  for (int p = t; p < 80; p += 256) {
    float v = 0.f;
    if (p < 36) { int i = p / 6, j = p % 6;
      v = 0.5f * (H0re[i * 6 + j] + H0re[j * 6 + i]); }
    else if (p < 72) { int q = p - 36; int i = q / 6, j = q % 6;
      v = 0.5f * (H0im[i * 6 + j] - H0im[j * 6 + i]); }
    H0f[p] = v;
  }
  for (int idx = t; idx < 72 * 32; idx += 256) {
    int p = idx / 32, k = idx % 32;
    float v;
    if (p < 36) { int i = p / 6, j = p % 6;
      v = 0.5f * (Sre[k * 36 + i * 6 + j] + Sre[k * 36 + j * 6 + i]); }
    else { int q = p - 36; int i = q / 6, j = q % 6;
      v = 0.5f * (Sim[k * 36 + i * 6 + j] - Sim[k * 36 + j * 6 + i]); }
    W2[idx] = v;
  }
}

// ---------------------------------------------------------------------------
// One complex-Hermitian Jacobi rotation zeroing A[P][Q]; compile-time P,Q keep
// all array indices constant so A/V stay in VGPRs.
// ---------------------------------------------------------------------------
template <int P, int Q>
__device__ __forceinline__ void jrot(float (&Ar)[6][6], float (&Ai)[6][6],
                                     float (&Vr)[6][6], float (&Vi)[6][6]) {
  float hre = Ar[P][Q], him = Ai[P][Q];
  float h2 = hre * hre + him * him;
  if (h2 <= 1e-30f) return;
  float ah  = sqrtf(h2);
  float a   = Ar[P][P], b = Ar[Q][Q];
  float tau = (a - b) / (2.f * ah);
  float t   = copysignf(1.f, tau) / (fabsf(tau) + sqrtf(1.f + tau * tau));
  float c   = 1.f / sqrtf(1.f + t * t);
  float sg  = t * c;
  float inv = 1.f / ah;
  float sr = sg * hre * inv, si = sg * him * inv;   // s = sg * h/|h|
#pragma unroll
  for (int j = 0; j < 6; ++j) {
    if (j == P || j == Q) continue;
    float xr = Ar[P][j], xi = Ai[P][j], yr = Ar[Q][j], yi = Ai[Q][j];
    float nxr = c * xr + (sr * yr - si * yi);       // row P' = c*rowP + s*rowQ
    float nxi = c * xi + (sr * yi + si * yr);
    float nyr = c * yr - (sr * xr + si * xi);       // row Q' = -conj(s)*rowP + c*rowQ
    float nyi = c * yi - (sr * xi - si * xr);
    Ar[P][j] = nxr; Ai[P][j] = nxi; Ar[Q][j] = nyr; Ai[Q][j] = nyi;
    Ar[j][P] = nxr; Ai[j][P] = -nxi; Ar[j][Q] = nyr; Ai[j][Q] = -nyi;
  }
  Ar[P][P] = a + t * ah; Ar[Q][Q] = b - t * ah;
  Ai[P][P] = 0.f; Ai[Q][Q] = 0.f;
  Ar[P][Q] = 0.f; Ai[P][Q] = 0.f; Ar[Q][P] = 0.f; Ai[Q][P] = 0.f;
#pragma unroll
  for (int j = 0; j < 6; ++j) {                     // V' = V*J
    float xr = Vr[j][P], xi = Vi[j][P], yr = Vr[j][Q], yi = Vi[j][Q];
    Vr[j][P] = c * xr + (sr * yr + si * yi);        // colP' = c*colP + conj(s)*colQ
    Vi[j][P] = c * xi + (sr * yi - si * yr);
    Vr[j][Q] = c * yr - (sr * xr - si * xi);        // colQ' = -s*colP + c*colQ
    Vi[j][Q] = c * yi - (sr * xi + si * xr);
  }
}

// ---------------------------------------------------------------------------
// Main kernel: 128 threads = 4 wave32; each wave owns 32 batch rows.
// GEMM1 (WMMA f32 16x16x4) -> LDS -> per-lane Jacobi -> projector features ->
// GEMM2 (WMMA f32 16x16x4) -> out + bias.
// ---------------------------------------------------------------------------
#define LSTRIDE 84  // >= 80 so full 16-wide tiles store unguarded; staggers banks

__launch_bounds__(128, 1)
__global__ void UnitaryPMM_main(const float* __restrict__ x, const float* __restrict__ Bg,
                                const float* __restrict__ H0f, const float* __restrict__ W2,
                                const float* __restrict__ bias, float* __restrict__ out) {
  __shared__ float lds[128 * LSTRIDE];
  const int tid  = threadIdx.x;
  const int wave = tid >> 5, lane = tid & 31;
  const int lo = lane & 15, hi = lane >> 4;
  const int wRow  = wave * 32;
  const int bBase = blockIdx.x * 128;

  // ---------------- GEMM1: Hfeat(32x72) = x_tile(32x64) @ Bg(64x80) --------
  v2f afrag[2][16];
#pragma unroll
  for (int rt = 0; rt < 2; ++rt) {
    const float* xp = x + (size_t)(bBase + wRow + rt * 16 + lo) * 64 + hi * 2;
#pragma unroll
    for (int kb = 0; kb < 16; ++kb)
      afrag[rt][kb] = *(const v2f*)(xp + kb * 4);
  }

#pragma unroll 1
  for (int nt = 0; nt < 5; ++nt) {
    v2f bfrag[16];
#pragma unroll
    for (int kb = 0; kb < 16; ++kb) {
      const float* bp = Bg + (kb * 4 + hi * 2) * 80 + nt * 16 + lo;
      bfrag[kb].x = bp[0];
      bfrag[kb].y = bp[80];
    }
#pragma unroll
    for (int rt = 0; rt < 2; ++rt) {
      v8f cc = {};
#pragma unroll
      for (int kb = 0; kb < 16; ++kb)
        cc = __builtin_amdgcn_wmma_f32_16x16x4_f32(false, afrag[rt][kb],
                                                   false, bfrag[kb],
                                                   (short)0, cc, false, false);
      // Unguarded store of all 16 columns: cols 72..79 land in row padding
      // (LSTRIDE=84), nothing reads them -> EXEC stays all-1s through GEMM1.
#pragma unroll
      for (int r = 0; r < 8; ++r) {
        int m = wRow + rt * 16 + hi * 8 + r;
        lds[m * LSTRIDE + nt * 16 + lo] = cc[r];
      }
    }
  }
  __syncthreads();

  // ---------------- per-lane 6x6 complex Hermitian eigensolver -------------
  {
    float Ar[6][6], Ai[6][6], Vr[6][6], Vi[6][6];
    const float* f = &lds[tid * LSTRIDE];
#pragma unroll
    for (int i = 0; i < 6; ++i)
#pragma unroll
      for (int j = 0; j < 6; ++j) {
        Ar[i][j] = f[i * 6 + j] + H0f[i * 6 + j];
        Ai[i][j] = f[36 + i * 6 + j] + H0f[36 + i * 6 + j];
        Vr[i][j] = (i == j) ? 1.f : 0.f;
        Vi[i][j] = 0.f;
      }
#pragma unroll 1
    for (int sweep = 0; sweep < 7; ++sweep) {
      jrot<0, 1>(Ar, Ai, Vr, Vi); jrot<0, 2>(Ar, Ai, Vr, Vi);
      jrot<0, 3>(Ar, Ai, Vr, Vi); jrot<0, 4>(Ar, Ai, Vr, Vi);
      jrot<0, 5>(Ar, Ai, Vr, Vi); jrot<1, 2>(Ar, Ai, Vr, Vi);
      jrot<1, 3>(Ar, Ai, Vr, Vi); jrot<1, 4>(Ar, Ai, Vr, Vi);
      jrot<1, 5>(Ar, Ai, Vr, Vi); jrot<2, 3>(Ar, Ai, Vr, Vi);
      jrot<2, 4>(Ar, Ai, Vr, Vi); jrot<2, 5>(Ar, Ai, Vr, Vi);
      jrot<3, 4>(Ar, Ai, Vr, Vi); jrot<3, 5>(Ar, Ai, Vr, Vi);
      jrot<4, 5>(Ar, Ai, Vr, Vi);
    }
    // select 3 smallest eigenvalues via rank (basis-free: projector below)
    float w[6];
#pragma unroll
    for (int m = 0; m < 6; ++m) {
      int rank = 0;
#pragma unroll
      for (int n = 0; n < 6; ++n)
        rank += (Ar[n][n] < Ar[m][m]) || (Ar[n][n] == Ar[m][m] && n < m);
      w[m] = (rank < 3) ? 1.f : 0.f;
    }
    // projector G = sum_m w_m v_m v_m^H  ->  features (Gre | Gim) into LDS row
    float* g = &lds[tid * LSTRIDE];
#pragma unroll
    for (int i = 0; i < 6; ++i)
#pragma unroll
      for (int j = 0; j < 6; ++j) {
        float gr = 0.f, gi = 0.f;
#pragma unroll
        for (int m = 0; m < 6; ++m) {
          gr += w[m] * (Vr[i][m] * Vr[j][m] + Vi[i][m] * Vi[j][m]);
          gi += w[m] * (Vi[i][m] * Vr[j][m] - Vr[i][m] * Vi[j][m]);
        }
        g[i * 6 + j] = gr;
        g[36 + i * 6 + j] = gi;
      }
  }
  __syncthreads();

  // ---------------- GEMM2: out(32x32) = f(32x72) @ W2(72x32) + bias --------
  // Hoist all B fragments (2 col-tiles x 18 k-steps) -- loaded once per wave.
  v2f b2frag[2][18];
#pragma unroll
  for (int nt = 0; nt < 2; ++nt)
#pragma unroll
    for (int kb = 0; kb < 18; ++kb) {
      const float* wp = W2 + (kb * 4 + hi * 2) * 32 + nt * 16 + lo;
      b2frag[nt][kb].x = wp[0];
      b2frag[nt][kb].y = wp[32];
    }
  const float bv0 = bias[lo];
  const float bv1 = bias[16 + lo];

#pragma unroll 1
  for (int rt = 0; rt < 2; ++rt) {
    v2f a2[18];
    const float* fp = &lds[(wRow + rt * 16 + lo) * LSTRIDE + hi * 2];
#pragma unroll
    for (int kb = 0; kb < 18; ++kb)
      a2[kb] = *(const v2f*)(fp + kb * 4);
#pragma unroll
    for (int nt = 0; nt < 2; ++nt) {
      v8f cc = {};
#pragma unroll
      for (int kb = 0; kb < 18; ++kb)
        cc = __builtin_amdgcn_wmma_f32_16x16x4_f32(false, a2[kb], false,
                                                   b2frag[nt][kb],
                                                   (short)0, cc, false, false);
      float bv = nt ? bv1 : bv0;
#pragma unroll
      for (int r = 0; r < 8; ++r) {
        int m = bBase + wRow + rt * 16 + hi * 8 + r;
        out[(size_t)m * 32 + nt * 16 + lo] = cc[r] + bv;
      }
    }
  }
}

extern "C" void kernel_launch(void* const* d_in, const int* in_sizes, int n_in,
                              void* d_out, int out_size, void* d_ws, size_t ws_size,
                              hipStream_t stream) {
  const float* x    = (const float*)d_in[0];
  const float* H0re = (const float*)d_in[1];
  const float* H0im = (const float*)d_in[2];
  const float* Hlre = (const float*)d_in[3];
  const float* Hlim = (const float*)d_in[4];
  const float* Sre  = (const float*)d_in[5];
  const float* Sim  = (const float*)d_in[6];
  const float* bias = (const float*)d_in[7];
  float* out = (float*)d_out;

  float* ws  = (float*)d_ws;
  float* Bg  = ws;              // 64*80 = 5120 floats
  float* H0f = Bg + 64 * 80;    // 80 floats
  float* W2  = H0f + 80;        // 72*32 = 2304 floats   (total ~30 KB)

  UnitaryPMM_prep<<<1, 256, 0, stream>>>(H0re, H0im, Hlre, Hlim, Sre, Sim, Bg, H0f, W2);

  // 131072 batch / 128 per block = 1024 blocks (exact, no tail -> EXEC all-1s
  // at every WMMA as the ISA requires)
  UnitaryPMM_main<<<1024, 128, 0, stream>>>(x, Bg, H0f, W2, bias, out);
}